// DiamondLayer_26792005992502
// MI455X (gfx1250) — compile-verified
//
#include <hip/hip_runtime.h>

// DiamondLayer: out[b,d] = mean over 16x16 window x[b, d+i, d+17+j]
// x: (32, 2048, 2048) f32, out: (32, 2016) f32
#define MATSIZE 2048
#define DSIZE   16
#define ND      (MATSIZE - 2 * DSIZE)   // 2016

typedef __attribute__((ext_vector_type(2))) float v2f;
typedef __attribute__((ext_vector_type(4))) float v4f;
typedef __attribute__((ext_vector_type(8))) float v8f;

__global__ __launch_bounds__(32)
void diamond_wmma_kernel(const float* __restrict__ x, float* __restrict__ out) {
    const int d0   = blockIdx.x * 16;      // output tile: d0 .. d0+15
    const int b    = blockIdx.y;           // batch
    const int lane = threadIdx.x;          // 0..31 (one wave32)
    const int half = lane >> 4;            // 0 or 1 (half-wave)
    const int lm   = lane & 15;            // M / N / T index within half

    // Patch of x needed by this tile: rows d0..d0+31, cols d0+16..d0+47.
    // Row stride padded to 36 dwords: 144B (16B aligned), bank-stride 36 (odd*4,
    // coprime spread over 64 banks for column-strided reads).
    __shared__ float patch[32][36];
    __shared__ float H[32][17];            // H[r][t], r in 0..31, t in 0..15

    // ---- coalesced-per-row global load: lane L fetches row L as 8x float4 ----
    const float* src = x + (size_t)b * MATSIZE * MATSIZE
                         + (size_t)d0 * MATSIZE + (size_t)(d0 + 16);
    {
        const float* rowp = src + (size_t)lane * MATSIZE;
#pragma unroll
        for (int q = 0; q < 8; ++q) {
            v4f v = *(const v4f*)(rowp + q * 4);
            *(v4f*)&patch[lane][q * 4] = v;
        }
    }
    __syncthreads();

    // ---- Stage 1 (WMMA, exact f32):  H = P(16x32) x W(32x16) per row-tile ----
    // W[k][t] = 1  iff  1 <= k - t <= 16   =>  H[r][t] = sum_{j=0..15} patch[r][t+1+j]
    v8f acc0 = {};   // rows 0..15
    v8f acc1 = {};   // rows 16..31
#pragma unroll
    for (int k0 = 0; k0 < 32; k0 += 4) {
        const int ka = k0 + half * 2;      // this lane's two K indices: ka, ka+1
        // A fragments (16x4 f32): lane<16 -> K={k0,k0+1}, lane>=16 -> K={k0+2,k0+3}
        v2f a0, a1;
        a0.x = patch[lm][ka];        a0.y = patch[lm][ka + 1];
        a1.x = patch[16 + lm][ka];   a1.y = patch[16 + lm][ka + 1];
        // B fragment (4x16 f32): VGPR0 -> row ka, VGPR1 -> row ka+1, N = lm
        const int dk = ka - lm;
        v2f bw;
        bw.x = (dk >= 1 && dk <= 16) ? 1.0f : 0.0f;   // row ka
        bw.y = (dk >= 0 && dk <= 15) ? 1.0f : 0.0f;   // row ka+1
        acc0 = __builtin_amdgcn_wmma_f32_16x16x4_f32(
                   /*neg_a=*/false, a0, /*neg_b=*/false, bw,
                   /*c_mod=*/(short)0, acc0, /*reuse_a=*/false, /*reuse_b=*/false);
        acc1 = __builtin_amdgcn_wmma_f32_16x16x4_f32(
                   /*neg_a=*/false, a1, /*neg_b=*/false, bw,
                   /*c_mod=*/(short)0, acc1, /*reuse_a=*/false, /*reuse_b=*/false);
    }

    // ---- spill H to LDS. C/D layout: VGPR v -> (M = v + 8*half, N = lm) ----
#pragma unroll
    for (int v = 0; v < 8; ++v) {
        const int m = v + half * 8;
        H[m][lm]      = acc0[v];
        H[16 + m][lm] = acc1[v];
    }
    __syncthreads();

    // ---- Stage 2: out[d0+t] = (1/256) * sum_{i=0..15} H[t+i][t] ----
    if (lane < 16) {
        float s = 0.0f;
#pragma unroll
        for (int i = 0; i < 16; ++i) s += H[lane + i][lane];
        out[(size_t)b * ND + d0 + lane] = s * (1.0f / 256.0f);
    }
}

extern "C" void kernel_launch(void* const* d_in, const int* in_sizes, int n_in,
                              void* d_out, int out_size, void* d_ws, size_t ws_size,
                              hipStream_t stream) {
    (void)in_sizes; (void)n_in; (void)out_size; (void)d_ws; (void)ws_size;
    const float* x = (const float*)d_in[0];
    float* out = (float*)d_out;
    dim3 grid(ND / 16, 32);   // 126 d-tiles x 32 batches
    diamond_wmma_kernel<<<grid, 32, 0, stream>>>(x, out);
}